// seir_base_35777077576087
// MI455X (gfx1250) — compile-verified
//
#include <hip/hip_runtime.h>

typedef __attribute__((ext_vector_type(2))) float v2f;
typedef __attribute__((ext_vector_type(8))) float v8f;

#define WAVES_PER_BLOCK 8
#define XSTRIDE 164                    // padded per-region LDS stride (floats) -> bank-conflict-free b64 reads
#define XFLOATS (16 * XSTRIDE)         // 2624 floats
#define APFLOATS 400                   // 16 regions * 25
#define WAVELDS (XFLOATS + APFLOATS)   // 3024 floats = 12096 B (16B aligned)

__device__ __forceinline__ float bcast_f(float x) {
    // force wave-uniform value into an SGPR (int-typed readfirstlane: portable)
    return __uint_as_float(__builtin_amdgcn_readfirstlane(__float_as_uint(x)));
}

// out[n,i,o] = relu( sum_c ( sum_j A[i,j]*Ap[n,i,j]*x[n,j,c] ) * fc_w[o,c] + fc_b[o] )
__global__ __launch_bounds__(WAVES_PER_BLOCK * 32)
void seir_fused_wmma(const float* __restrict__ A5,   // (5,5)
                     const float* __restrict__ Ap,   // (N,5,5)
                     const float* __restrict__ X,    // (N,5,32)
                     const float* __restrict__ FW,   // (32,32) torch Linear weight (o,c)
                     const float* __restrict__ FB,   // (32,)
                     float* __restrict__ Out,        // (N,5,32)
                     int nReg)
{
    __shared__ float lds[WAVES_PER_BLOCK * WAVELDS];
    const int lane = threadIdx.x & 31;
    const int wv   = threadIdx.x >> 5;
    const int r16  = lane & 15;   // region-in-tile (A: M index, B/D: N index)
    const int hl   = lane >> 4;   // lane half -> selects K pair / M offset 8
    float* ldsX  = lds + wv * WAVELDS;
    float* ldsAp = ldsX + XFLOATS;

    // ---- uniform 5x5 A pinned to SGPRs (cuts VGPR pressure below the 256 boundary) ----
    float a5[25];
#pragma unroll
    for (int t = 0; t < 25; ++t) a5[t] = bcast_f(A5[t]);

    // ---- resident B fragments: B[k][n] = fc_w[n][kGlobal]; lane holds N=r16, K=2*hl+{0,1} ----
    v2f bf[2][8];
#pragma unroll
    for (int nt = 0; nt < 2; ++nt)
#pragma unroll
        for (int kk = 0; kk < 8; ++kk)
            bf[nt][kk] = *reinterpret_cast<const v2f*>(FW + (nt * 16 + r16) * 32 + kk * 4 + 2 * hl);

    float bias[2];
    bias[0] = FB[r16];
    bias[1] = FB[16 + r16];

    const int tiles = (nReg + 15) >> 4;
    const int gw = blockIdx.x * WAVES_PER_BLOCK + wv;
    const int nw = gridDim.x * WAVES_PER_BLOCK;

    for (int tile = gw; tile < tiles; tile += nw) {   // wave-uniform loop: EXEC stays all-1s
        const int rbase = tile << 4;
        int rcount = nReg - rbase; if (rcount > 16) rcount = 16;

        // ---- async global->LDS fill: no VGPR staging, tracked by ASYNCcnt ----
        {
            const float* gx = X + (size_t)rbase * 160;
            const int nv4 = rcount * 40;              // float4 count (160 floats/region)
            for (int it = lane; it < nv4; it += 32) {
                int g   = it * 4;
                int r   = g / 160;                    // 160 % 4 == 0 -> b128 never crosses region
                int rem = g - r * 160;
                unsigned int  la = (unsigned int)(uintptr_t)(ldsX + r * XSTRIDE + rem);
                unsigned long long ga = (unsigned long long)(uintptr_t)(gx + g);
                asm volatile("global_load_async_to_lds_b128 %0, %1, off"
                             :: "v"(la), "v"(ga) : "memory");
            }
            const float* gap = Ap + (size_t)rbase * 25;
            for (int it = lane; it < rcount * 25; it += 32) {
                unsigned int  la = (unsigned int)(uintptr_t)(ldsAp + it);
                unsigned long long ga = (unsigned long long)(uintptr_t)(gap + it);
                asm volatile("global_load_async_to_lds_b32 %0, %1, off"
                             :: "v"(la), "v"(ga) : "memory");
            }
        }
        // async writes visible in LDS once ASYNCcnt drains; clobber blocks DS-read hoisting
        asm volatile("s_wait_asynccnt 0x0" ::: "memory");

        // ---- per-lane region weights W[i][j] = A[i][j] * Ap[r,i,j] ----
        float wreg[25];
#pragma unroll
        for (int t = 0; t < 25; ++t) wreg[t] = a5[t] * ldsAp[r16 * 25 + t];

        // ---- accumulators preloaded with bias: C[m][n] = b[n] ----
        v8f acc[5][2];
#pragma unroll
        for (int i = 0; i < 5; ++i)
#pragma unroll
            for (int nt = 0; nt < 2; ++nt) {
                float b = bias[nt];
                v8f c0 = {b, b, b, b, b, b, b, b};
                acc[i][nt] = c0;
            }

        // ---- K loop: 8 steps of 16x16x4 fp32 WMMA, A fragments built with packed FMAs ----
#pragma unroll
        for (int kk = 0; kk < 8; ++kk) {
            const int c = kk * 4 + 2 * hl;            // lane's two channels this K-step
            v2f xv[5];
#pragma unroll
            for (int j = 0; j < 5; ++j)
                xv[j] = *reinterpret_cast<const v2f*>(ldsX + r16 * XSTRIDE + j * 32 + c);
#pragma unroll
            for (int i = 0; i < 5; ++i) {
                v2f h = wreg[i * 5 + 0] * xv[0];
                h += wreg[i * 5 + 1] * xv[1];
                h += wreg[i * 5 + 2] * xv[2];
                h += wreg[i * 5 + 3] * xv[3];
                h += wreg[i * 5 + 4] * xv[4];
                acc[i][0] = __builtin_amdgcn_wmma_f32_16x16x4_f32(
                    false, h, false, bf[0][kk], (short)0, acc[i][0], false, false);
                acc[i][1] = __builtin_amdgcn_wmma_f32_16x16x4_f32(
                    false, h, false, bf[1][kk], (short)0, acc[i][1], false, false);
            }
        }

        // ---- ReLU + store; D lane L, VGPR d -> region d+8*hl, channel nt*16+r16 ----
#pragma unroll
        for (int i = 0; i < 5; ++i)
#pragma unroll
            for (int nt = 0; nt < 2; ++nt)
#pragma unroll
                for (int d = 0; d < 8; ++d) {
                    int r = d + 8 * hl;
                    if (r < rcount) {
                        float vv = fmaxf(acc[i][nt][d], 0.0f);
                        Out[(size_t)(rbase + r) * 160 + i * 32 + nt * 16 + r16] = vv;
                    }
                }
        // WAR: all DS reads of this slice must retire before next tile's async fill lands
        asm volatile("s_wait_dscnt 0x0" ::: "memory");
    }
}

extern "C" void kernel_launch(void* const* d_in, const int* in_sizes, int n_in,
                              void* d_out, int out_size, void* d_ws, size_t ws_size,
                              hipStream_t stream) {
    const float* A5 = (const float*)d_in[0];   // (5,5)
    const float* Ap = (const float*)d_in[1];   // (N,5,5)
    const float* X  = (const float*)d_in[2];   // (N,5,32)
    const float* FW = (const float*)d_in[3];   // (32,32)
    const float* FB = (const float*)d_in[4];   // (32,)
    float* Out = (float*)d_out;

    int nReg  = in_sizes[1] / 25;
    int tiles = (nReg + 15) / 16;
    int blocks = (tiles + WAVES_PER_BLOCK - 1) / WAVES_PER_BLOCK;
    if (blocks > 1024) blocks = 1024;          // grid-stride loop covers the rest
    if (blocks < 1) blocks = 1;

    seir_fused_wmma<<<blocks, WAVES_PER_BLOCK * 32, 0, stream>>>(A5, Ap, X, FW, FB, Out, nReg);
}